// MultiClassBoundingBoxRegressor_37237366456337
// MI455X (gfx1250) — compile-verified
//
#include <hip/hip_runtime.h>
#include <math.h>

// ---------------------------------------------------------------------------
// MultiClassBoundingBoxRegressor fused kernel for gfx1250 (MI455X, wave32)
//   coords/presence projection via V_WMMA_F32_16X16X4_F32, then per-(b,c)
//   sigmoid -> top-K -> NMS -> masked outputs, all in one workgroup pass.
// ---------------------------------------------------------------------------

typedef float v2f __attribute__((ext_vector_type(2)));
typedef float v8f __attribute__((ext_vector_type(8)));

namespace {
constexpr int Bn = 32;    // batch
constexpr int Cn = 30;    // classes
constexpr int Rn = 196;   // regions (14x14 grid)
constexpr int Dn = 256;   // feature dim
constexpr int Kn = 20;    // top-k
constexpr int Gn = 14;    // grid side
constexpr float CONF = 0.5f;
constexpr float IOU_THR = 0.3f;

// d_out layout (flat f32, reference tuple order):
constexpr int OFF_COORDS = 0;                       // (B,C,K,4)
constexpr int OFF_PROBS  = Bn * Cn * Kn * 4;        // (B,C,K)
constexpr int OFF_KEEP   = OFF_PROBS + Bn * Cn * Kn;// (B,C,K)
constexpr int OFF_PRES   = OFF_KEEP + Bn * Cn * Kn; // (B,C,R,1)
}

__global__ __launch_bounds__(256)
void bbox_regressor_fused(const float* __restrict__ lf,       // (B,C,R,D)
                          const float* __restrict__ Wc,       // (4,D)
                          const float* __restrict__ bc_bias,  // (4,)
                          const float* __restrict__ Wp,       // (1,D)
                          const float* __restrict__ bp_bias,  // (1,)
                          float* __restrict__ out) {
  __shared__ float wpad[16 * Dn];          // zero-padded [16][256] weight block
  __shared__ float sCoords[Rn * 4];        // final coords (bias + grid center)
  __shared__ float sProbs[Rn];             // sigmoid(presence)

  const int tid  = threadIdx.x;
  const int lane = tid & 31;
  const int wave = tid >> 5;               // 8 waves of 32 (wave32)
  const int bcid = blockIdx.x;             // 0 .. B*C-1

  // ---- stage padded weights: wpad[n][k] = n<4 ? Wc[n][k] : n==4 ? Wp[k] : 0
  for (int i = tid; i < 16 * Dn; i += 256) {
    const int n = i >> 8;
    const int k = i & (Dn - 1);
    float v = 0.0f;
    if (n < 4)       v = Wc[n * Dn + k];
    else if (n == 4) v = Wp[k];
    wpad[i] = v;
  }
  __syncthreads();

  // ---- Phase A: GEMM (196 x 256) x (256 x 16[5 used]) via WMMA f32 16x16x4
  const int m    = lane & 15;              // A row within tile / B column index
  const int half = lane >> 4;              // K sub-pair selector {0,1}
  const float* Abase = lf + (size_t)bcid * Rn * Dn;

  for (int t = wave; t < 13; t += 8) {     // 13 row-tiles of 16 cover 196 rows
    const int rowBase = t * 16;
    int r = rowBase + m;
    if (r > Rn - 1) r = Rn - 1;            // clamp tail reads (writes masked)
    const float* pa = Abase + (size_t)r * Dn + half * 2;
    const float* pb = &wpad[m * Dn + half * 2];

    v8f acc = {0.f, 0.f, 0.f, 0.f, 0.f, 0.f, 0.f, 0.f};
    #pragma unroll 4
    for (int kb = 0; kb < Dn; kb += 4) {
      v2f a = *reinterpret_cast<const v2f*>(pa + kb);   // global_load_b64
      v2f b = *reinterpret_cast<const v2f*>(pb + kb);   // ds_load_b64
      // (neg_a, A, neg_b, B, c_mod, C, reuse_a, reuse_b)
      acc = __builtin_amdgcn_wmma_f32_16x16x4_f32(
          false, a, false, b, (short)0, acc, false, false);
    }

    // C/D layout: VGPR j, lanes 0-15 -> row j, lanes 16-31 -> row j+8; col = lane%16
    const int col = m;
    if (col < 5) {
      #pragma unroll
      for (int j = 0; j < 8; ++j) {
        const int row = rowBase + j + half * 8;
        if (row < Rn) {
          const float v = acc[j];
          if (col == 4) {
            const float pres = v + bp_bias[0];
            out[OFF_PRES + (size_t)bcid * Rn + row] = pres;      // raw presence
            sProbs[row] = 1.0f / (1.0f + __expf(-pres));
          } else {
            const int gy = row / Gn, gx = row - gy * Gn;
            const float cx = (gx + 0.5f) * (1.0f / Gn);
            const float cy = (gy + 0.5f) * (1.0f / Gn);
            sCoords[row * 4 + col] =
                v + bc_bias[col] + ((col & 1) ? cy : cx);
          }
        }
      }
    }
  }
  __syncthreads();

  // ---- Phase B/C: top-K + NMS + masked writes (tiny; serial on thread 0)
  if (tid == 0) {
    float vals[Kn];
    float bx[Kn][4];
    bool  keep[Kn];

    for (int k = 0; k < Kn; ++k) {         // stable argmax == jax top_k order
      float best = -INFINITY;
      int bi = 0;
      for (int r2 = 0; r2 < Rn; ++r2) {
        const float p = sProbs[r2];
        if (p > best) { best = p; bi = r2; }
      }
      vals[k] = best;
      sProbs[bi] = -INFINITY;
      #pragma unroll
      for (int o = 0; o < 4; ++o) bx[k][o] = sCoords[bi * 4 + o];
      keep[k] = best > CONF;
    }

    float area[Kn];
    for (int k = 0; k < Kn; ++k) {
      const float w = fmaxf(bx[k][2] - bx[k][0], 0.0f);
      const float h = fmaxf(bx[k][3] - bx[k][1], 0.0f);
      area[k] = w * h;
    }
    for (int i = 0; i < Kn; ++i) {
      if (!keep[i]) continue;
      for (int j = i + 1; j < Kn; ++j) {
        if (!keep[j]) continue;
        const float xx1 = fmaxf(bx[i][0], bx[j][0]);
        const float yy1 = fmaxf(bx[i][1], bx[j][1]);
        const float xx2 = fminf(bx[i][2], bx[j][2]);
        const float yy2 = fminf(bx[i][3], bx[j][3]);
        const float inter = fmaxf(xx2 - xx1, 0.0f) * fmaxf(yy2 - yy1, 0.0f);
        const float uni = fmaxf(area[i] + area[j] - inter, 1e-9f);
        if (inter / uni > IOU_THR) keep[j] = false;
      }
    }

    const size_t cbase = (size_t)bcid * Kn * 4;
    const size_t kbase = (size_t)bcid * Kn;
    for (int k = 0; k < Kn; ++k) {
      const float f = keep[k] ? 1.0f : 0.0f;
      #pragma unroll
      for (int o = 0; o < 4; ++o)
        out[OFF_COORDS + cbase + k * 4 + o] = bx[k][o] * f;
      out[OFF_PROBS + kbase + k] = vals[k] * f;
      out[OFF_KEEP  + kbase + k] = f;
    }
  }
}

extern "C" void kernel_launch(void* const* d_in, const int* in_sizes, int n_in,
                              void* d_out, int out_size, void* d_ws, size_t ws_size,
                              hipStream_t stream) {
  (void)in_sizes; (void)n_in; (void)d_ws; (void)ws_size; (void)out_size;
  const float* lf = (const float*)d_in[0];
  const float* Wc = (const float*)d_in[1];
  const float* bc = (const float*)d_in[2];
  const float* Wp = (const float*)d_in[3];
  const float* bp = (const float*)d_in[4];
  float* out = (float*)d_out;

  bbox_regressor_fused<<<dim3(Bn * Cn), dim3(256), 0, stream>>>(
      lf, Wc, bc, Wp, bp, out);
}